// GateGCNPyG_51951924412559
// MI455X (gfx1250) — compile-verified
//
#include <hip/hip_runtime.h>
#include <math.h>

typedef __attribute__((ext_vector_type(2))) float v2f;
typedef __attribute__((ext_vector_type(8))) float v8f;

#define HD 256          // hidden/out feature width (all layers emit 256)
#define LN_EPS 1e-5f
#define NEG_SLOPE 0.2f

// ---------------------------------------------------------------------------
// GEMM: C[nrows x 256] = A[nrows x K] * W[K x 256] + bias[256]
// One wave computes a 16x64 tile (4 column sub-tiles of 16) using
// V_WMMA_F32_16X16X4_F32.  f32 A-fragment layout (ISA 7.12.2):
//   lanes 0-15  hold row M=lane,    VGPR0=K0, VGPR1=K1
//   lanes 16-31 hold row M=lane-16, VGPR0=K2, VGPR1=K3
// B-fragment mirrors this with K rows striped the same way, N = lane&15.
// C/D: VGPR r -> row (r + 8*half), col = lane&15.
// ---------------------------------------------------------------------------
template <int K>
__launch_bounds__(256)
__global__ void gemm_node_wmma(const float* __restrict__ A,
                               const float* __restrict__ W,
                               const float* __restrict__ bias,
                               float* __restrict__ C,
                               int nrows)
{
    const int lane = threadIdx.x & 31;
    const int warp = threadIdx.x >> 5;
    const int task = blockIdx.x * 8 + warp;   // wave-task: 16 rows x 64 cols
    const int rowTile = task >> 2;
    const int colGrp  = task & 3;
    if (rowTile * 16 >= nrows) return;
    const int row0 = rowTile * 16;
    const int col0 = colGrp * 64;

    const int half = lane >> 4;   // 0 -> K{0,1}, 1 -> K{2,3}
    const int m    = lane & 15;

    v8f acc0 = {}, acc1 = {}, acc2 = {}, acc3 = {};

    for (int k0 = 0; k0 < K; k0 += 4) {
        // A fragment (two consecutive f32 per lane)
        v2f a = *(const v2f*)(A + (size_t)(row0 + m) * K + k0 + 2 * half);
        // B fragments for the four 16-wide column tiles
        const float* Wk0 = W + (size_t)(k0 + 2 * half) * HD + col0 + m;
        const float* Wk1 = Wk0 + HD;
        v2f b0 = { Wk0[0],  Wk1[0]  };
        v2f b1 = { Wk0[16], Wk1[16] };
        v2f b2 = { Wk0[32], Wk1[32] };
        v2f b3 = { Wk0[48], Wk1[48] };
        acc0 = __builtin_amdgcn_wmma_f32_16x16x4_f32(false, a, false, b0, (short)0, acc0, false, false);
        acc1 = __builtin_amdgcn_wmma_f32_16x16x4_f32(false, a, false, b1, (short)0, acc1, false, false);
        acc2 = __builtin_amdgcn_wmma_f32_16x16x4_f32(false, a, false, b2, (short)0, acc2, false, false);
        acc3 = __builtin_amdgcn_wmma_f32_16x16x4_f32(false, a, false, b3, (short)0, acc3, false, false);
    }

    v8f accs[4] = { acc0, acc1, acc2, acc3 };
    #pragma unroll
    for (int c = 0; c < 4; ++c) {
        const int col = col0 + 16 * c + m;
        const float bv = bias[col];
        float* Cp = C + (size_t)(row0 + 8 * half) * HD + col;
        #pragma unroll
        for (int r = 0; r < 8; ++r) {
            Cp[(size_t)r * HD] = accs[c][r] + bv;
        }
    }
}

// ---------------------------------------------------------------------------
// Per-node attention scores: si = h . ai, sj = h . aj  (one wave per node)
// ---------------------------------------------------------------------------
__launch_bounds__(256)
__global__ void node_scores(const float* __restrict__ H,
                            const float* __restrict__ ai,
                            const float* __restrict__ aj,
                            float* __restrict__ si,
                            float* __restrict__ sj,
                            int n)
{
    const int node = (int)((blockIdx.x * blockDim.x + threadIdx.x) >> 5);
    const int lane = threadIdx.x & 31;
    if (node >= n) return;
    const float* h = H + (size_t)node * HD;
    float pi = 0.f, pj = 0.f;
    #pragma unroll
    for (int f = lane; f < HD; f += 32) {
        const float hv = h[f];
        pi += hv * ai[f];
        pj += hv * aj[f];
    }
    #pragma unroll
    for (int off = 16; off; off >>= 1) {
        pi += __shfl_xor(pi, off, 32);
        pj += __shfl_xor(pj, off, 32);
    }
    if (lane == 0) { si[node] = pi; sj[node] = pj; }
}

// ---------------------------------------------------------------------------
// Reset per-node segment state and the message accumulator.
// ---------------------------------------------------------------------------
__global__ void init_seg(float* __restrict__ acc,
                         float* __restrict__ mx,
                         float* __restrict__ den,
                         int n)
{
    const int i = blockIdx.x * blockDim.x + threadIdx.x;
    if (i < n * HD) acc[i] = 0.f;
    if (i < n) { mx[i] = -INFINITY; den[i] = 0.f; }
}

// ---------------------------------------------------------------------------
// Pass 1: logits + LeakyReLU + segment atomic-max (float max via bit trick).
// ---------------------------------------------------------------------------
__global__ void edge_logits(const int* __restrict__ src,
                            const int* __restrict__ tgt,
                            const float* __restrict__ si,
                            const float* __restrict__ sj,
                            const float* __restrict__ ab,
                            float* __restrict__ elog,
                            float* __restrict__ mx,
                            int ne)
{
    const int e = blockIdx.x * blockDim.x + threadIdx.x;
    if (e >= ne) return;
    const int s = src[e], t = tgt[e];
    float v = si[t] + sj[s] + ab[0];
    v = (v >= 0.f) ? v : NEG_SLOPE * v;
    elog[e] = v;
    if (v >= 0.f) atomicMax((int*)(mx + t), __float_as_int(v));
    else          atomicMin((unsigned int*)(mx + t), __float_as_uint(v));
}

// ---------------------------------------------------------------------------
// Pass 2: e = exp(logit - m[tgt]); denom[tgt] += e.
// ---------------------------------------------------------------------------
__global__ void edge_exp(const int* __restrict__ tgt,
                         const float* __restrict__ mx,
                         float* __restrict__ elog,
                         float* __restrict__ den,
                         int ne)
{
    const int e = blockIdx.x * blockDim.x + threadIdx.x;
    if (e >= ne) return;
    const int t = tgt[e];
    const float v = __expf(elog[e] - mx[t]);
    elog[e] = v;
    atomicAdd(den + t, v);
}

// ---------------------------------------------------------------------------
// Pass 3 (bandwidth critical): one wave per edge, 8 floats per lane (2xfloat4)
// msg = sigmoid(uh[tgt]+vh[src]) * alpha * h[src]; acc[tgt] += msg (f32 atomics)
// ---------------------------------------------------------------------------
__launch_bounds__(256)
__global__ void edge_messages(const int* __restrict__ src,
                              const int* __restrict__ tgt,
                              const float* __restrict__ elog,
                              const float* __restrict__ den,
                              const float* __restrict__ H,
                              const float* __restrict__ UH,
                              const float* __restrict__ VH,
                              float* __restrict__ acc,
                              int ne)
{
    const int e = (int)((blockIdx.x * blockDim.x + threadIdx.x) >> 5);
    const int lane = threadIdx.x & 31;
    if (e >= ne) return;
    const int s = src[e], t = tgt[e];
    const float alpha = elog[e] / (den[t] + 1e-16f);
    const int f0 = lane * 8;
    const float4* u4 = (const float4*)(UH + (size_t)t * HD + f0);
    const float4* v4 = (const float4*)(VH + (size_t)s * HD + f0);
    const float4* h4 = (const float4*)(H  + (size_t)s * HD + f0);
    float* ap = acc + (size_t)t * HD + f0;
    #pragma unroll
    for (int c = 0; c < 2; ++c) {
        const float4 u = u4[c], v = v4[c], h = h4[c];
        const float m0 = alpha * h.x / (1.f + __expf(-(u.x + v.x)));
        const float m1 = alpha * h.y / (1.f + __expf(-(u.y + v.y)));
        const float m2 = alpha * h.z / (1.f + __expf(-(u.z + v.z)));
        const float m3 = alpha * h.w / (1.f + __expf(-(u.w + v.w)));
        atomicAdd(ap + c * 4 + 0, m0);
        atomicAdd(ap + c * 4 + 1, m1);
        atomicAdd(ap + c * 4 + 2, m2);
        atomicAdd(ap + c * 4 + 3, m3);
    }
}

// ---------------------------------------------------------------------------
// Finalize: LayerNorm(acc) * g + b  (+ skip) (ReLU).  Block of 256 per node.
// Safe to run in place (out may alias acc).
// ---------------------------------------------------------------------------
__launch_bounds__(256)
__global__ void node_finalize(const float* __restrict__ acc,
                              const float* __restrict__ g,
                              const float* __restrict__ b,
                              const float* __restrict__ skip,
                              float* __restrict__ out,
                              int doRelu)
{
    __shared__ float red[8];
    const int node = blockIdx.x;
    const int f = threadIdx.x;
    const float v = acc[(size_t)node * HD + f];

    // mean
    float s = v;
    #pragma unroll
    for (int off = 16; off; off >>= 1) s += __shfl_xor(s, off, 32);
    if ((f & 31) == 0) red[f >> 5] = s;
    __syncthreads();
    float total = 0.f;
    #pragma unroll
    for (int i = 0; i < 8; ++i) total += red[i];
    const float mu = total * (1.f / HD);
    __syncthreads();

    // variance (population, ddof = 0)
    const float d = v - mu;
    float s2 = d * d;
    #pragma unroll
    for (int off = 16; off; off >>= 1) s2 += __shfl_xor(s2, off, 32);
    if ((f & 31) == 0) red[f >> 5] = s2;
    __syncthreads();
    float tot2 = 0.f;
    #pragma unroll
    for (int i = 0; i < 8; ++i) tot2 += red[i];
    const float var = tot2 * (1.f / HD);

    float y = d * rsqrtf(var + LN_EPS) * g[f] + b[f];
    if (skip) y += skip[(size_t)node * HD + f];
    if (doRelu) y = fmaxf(y, 0.f);
    out[(size_t)node * HD + f] = y;
}

// ---------------------------------------------------------------------------
// Host-side orchestration
// ---------------------------------------------------------------------------
struct ConvP {
    const float *W, *Wb, *U, *Ub, *V, *Vb, *ai, *aj, *ab, *g, *b;
};

static inline ConvP grab(void* const* d_in, int base) {
    ConvP p;
    p.W  = (const float*)d_in[base + 0];
    p.Wb = (const float*)d_in[base + 1];
    p.U  = (const float*)d_in[base + 2];
    p.Ub = (const float*)d_in[base + 3];
    p.V  = (const float*)d_in[base + 4];
    p.Vb = (const float*)d_in[base + 5];
    p.ai = (const float*)d_in[base + 6];
    p.aj = (const float*)d_in[base + 7];
    p.ab = (const float*)d_in[base + 8];
    p.g  = (const float*)d_in[base + 9];
    p.b  = (const float*)d_in[base + 10];
    return p;
}

template <int K>
static void run_layer(const float* xin, const ConvP& p,
                      const int* src, const int* tgt, int N, int NE,
                      float* H, float* UH, float* VH,
                      float* SI, float* SJ, float* MX, float* DEN, float* ELOG,
                      float* accBuf, const float* skip, float* out, int doRelu,
                      hipStream_t stream)
{
    const int gemmGrid = ((N / 16) * 4 + 7) / 8;
    gemm_node_wmma<K><<<gemmGrid, 256, 0, stream>>>(xin, p.W, p.Wb, H, N);
    gemm_node_wmma<HD><<<gemmGrid, 256, 0, stream>>>(H, p.U, p.Ub, UH, N);
    gemm_node_wmma<HD><<<gemmGrid, 256, 0, stream>>>(H, p.V, p.Vb, VH, N);

    node_scores<<<(N * 32 + 255) / 256, 256, 0, stream>>>(H, p.ai, p.aj, SI, SJ, N);
    init_seg<<<((size_t)N * HD + 255) / 256, 256, 0, stream>>>(accBuf, MX, DEN, N);
    edge_logits<<<(NE + 255) / 256, 256, 0, stream>>>(src, tgt, SI, SJ, p.ab, ELOG, MX, NE);
    edge_exp<<<(NE + 255) / 256, 256, 0, stream>>>(tgt, MX, ELOG, DEN, NE);
    edge_messages<<<((size_t)NE * 32 + 255) / 256, 256, 0, stream>>>(
        src, tgt, ELOG, DEN, H, UH, VH, accBuf, NE);
    node_finalize<<<N, 256, 0, stream>>>(accBuf, p.g, p.b, skip, out, doRelu);
}

extern "C" void kernel_launch(void* const* d_in, const int* in_sizes, int n_in,
                              void* d_out, int out_size, void* d_ws, size_t ws_size,
                              hipStream_t stream)
{
    const int N  = in_sizes[0] / 128;   // 50000
    const int NE = in_sizes[1] / 2;     // 800000

    const float* x   = (const float*)d_in[0];
    const int*   src = (const int*)d_in[1];
    const int*   tgt = src + NE;

    const ConvP l0 = grab(d_in, 2);
    const ConvP l1 = grab(d_in, 13);
    const ConvP l2 = grab(d_in, 24);
    const float* skipW = (const float*)d_in[35];
    const float* skipb = (const float*)d_in[36];

    // Workspace layout (floats)
    float* ws = (float*)d_ws;
    const size_t NH = (size_t)N * HD;
    float* H    = ws;
    float* UH   = ws + NH;
    float* VH   = ws + 2 * NH;
    float* BUFA = ws + 3 * NH;
    float* BUFB = ws + 4 * NH;
    float* SI   = ws + 5 * NH;
    float* SJ   = SI + N;
    float* MX   = SJ + N;
    float* DEN  = MX + N;
    float* ELOG = DEN + N;

    // x_res = x @ skipW + skipb  -> BUFB
    const int gemmGrid = ((N / 16) * 4 + 7) / 8;
    gemm_node_wmma<128><<<gemmGrid, 256, 0, stream>>>(x, skipW, skipb, BUFB, N);

    // layer0: input x (K=128), acc=BUFA, skip=x_res(BUFB), ReLU, out in place (BUFA)
    run_layer<128>(x, l0, src, tgt, N, NE, H, UH, VH, SI, SJ, MX, DEN, ELOG,
                   BUFA, BUFB, BUFA, 1, stream);
    // layer1: input BUFA (K=256), acc=BUFB, identity skip (BUFA), ReLU, out BUFB
    run_layer<HD>(BUFA, l1, src, tgt, N, NE, H, UH, VH, SI, SJ, MX, DEN, ELOG,
                  BUFB, BUFA, BUFB, 1, stream);
    // layer2: input BUFB, acc=BUFA, no skip, no ReLU, out -> d_out
    run_layer<HD>(BUFB, l2, src, tgt, N, NE, H, UH, VH, SI, SJ, MX, DEN, ELOG,
                  BUFA, nullptr, (float*)d_out, 0, stream);
}